// MSIW_73753178407365
// MI455X (gfx1250) — compile-verified
//
#include <hip/hip_runtime.h>
#include <math.h>

// Problem constants (from reference: nw_out [4, 19, 512, 1024] fp32)
#define N_      4
#define C_      19
#define H_      512
#define W_      1024
#define HW_     (H_ * W_)        // 524288 pixels per (n, c) plane
#define NP_     (N_ * HW_)       // 2097152 pixels total
#define BLOCK_  256              // 8 wave32 per workgroup
#define WAVES_  (BLOCK_ / 32)
#define NGROUPS_ (NP_ / 4)       // each "group" = 4 consecutive-w pixels (one float4 per channel)
#define GRID_   512              // pass-1 blocks; each thread loops NGROUPS_/(GRID_*BLOCK_) = 4 times
#define IW_     0.2f

// ---------------------------------------------------------------------------
// Workspace layout (d_ws): float classSum[19]; unsigned int classCount[19]
// ---------------------------------------------------------------------------
__global__ void msiw_zero_ws(float* ws) {
  unsigned int i = threadIdx.x;
  if (i < 2u * C_) ws[i] = 0.0f;  // bit pattern 0 zeroes the uint counts too
}

// ---------------------------------------------------------------------------
// Pass 1: single streaming sweep of the 159 MB tensor.
// Per pixel: argmax over C, s = sum_c prob_c^2 = (sum e^2) / (sum e)^2,
// accumulate {s, 1} into per-class bins (LDS per-wave copies -> global atomics).
// ---------------------------------------------------------------------------
__global__ __launch_bounds__(BLOCK_) void msiw_pass1(
    const float* __restrict__ x,
    float* __restrict__ gsum,
    unsigned int* __restrict__ gcnt) {
  __shared__ float        lsum[WAVES_][C_];
  __shared__ unsigned int lcnt[WAVES_][C_];
  const int tid = threadIdx.x;
  const int wv  = tid >> 5;  // wave32 id within the workgroup

  for (int i = tid; i < WAVES_ * C_; i += BLOCK_) {
    (&lsum[0][0])[i] = 0.0f;
    (&lcnt[0][0])[i] = 0u;
  }
  __syncthreads();

  const int stride = GRID_ * BLOCK_;  // groups consumed per grid sweep
  for (int g = blockIdx.x * BLOCK_ + tid; g < NGROUPS_; g += stride) {
    const int pix = g << 2;                  // first of 4 pixels in this group
    const int n   = pix / HW_;
    const int rem = pix - n * HW_;
    const float* base = x + (size_t)n * (C_ * HW_) + rem;

    // 19 coalesced float4 loads (global_load_b128), one per channel
    float4 v[C_];
    #pragma unroll
    for (int c = 0; c < C_; ++c) {
      v[c] = *(const float4*)(base + (size_t)c * HW_);
    }

    // Prefetch next grid-stride tile (global_prefetch_b8; no LOADcnt cost)
    const int g2 = g + stride;
    if (g2 < NGROUPS_) {
      const int pix2 = g2 << 2;
      const int n2   = pix2 / HW_;
      const int rem2 = pix2 - n2 * HW_;
      const float* base2 = x + (size_t)n2 * (C_ * HW_) + rem2;
      #pragma unroll
      for (int c = 0; c < C_; ++c) {
        __builtin_prefetch(base2 + (size_t)c * HW_, 0, 0);
      }
    }

    // Process the 4 pixels held in the float4 lanes
    #pragma unroll
    for (int k = 0; k < 4; ++k) {
      float m = -3.402823466e38f;
      int   a = 0;
      #pragma unroll
      for (int c = 0; c < C_; ++c) {
        const float xv = ((const float*)&v[c])[k];
        if (xv > m) { m = xv; a = c; }   // strict '>' keeps first max (jnp.argmax)
      }
      float S = 0.0f, S2 = 0.0f;
      #pragma unroll
      for (int c = 0; c < C_; ++c) {
        const float e = __expf(((const float*)&v[c])[k] - m);
        S  += e;
        S2 += e * e;
      }
      atomicAdd(&lsum[wv][a], S2 / (S * S));  // ds_add_f32
      atomicAdd(&lcnt[wv][a], 1u);            // ds_add_u32
    }
  }

  __syncthreads();
  if (tid < C_) {
    float        s  = 0.0f;
    unsigned int ct = 0u;
    #pragma unroll
    for (int w = 0; w < WAVES_; ++w) { s += lsum[w][tid]; ct += lcnt[w][tid]; }
    atomicAdd(&gsum[tid], s);   // global_atomic_add_f32
    atomicAdd(&gcnt[tid], ct);  // global_atomic_add_u32
  }
}

// ---------------------------------------------------------------------------
// Pass 2 (tiny): den_c = max(hist^0.2 * Np^0.8, 1); out = -sum_k S_k/den_k / (N*C)
// ---------------------------------------------------------------------------
__global__ void msiw_finalize(const float* __restrict__ gsum,
                              const unsigned int* __restrict__ gcnt,
                              float* __restrict__ out) {
  __shared__ float partial[C_];
  const int t = threadIdx.x;
  if (t < C_) {
    const float hist = (float)gcnt[t];
    float den = powf(hist, IW_) * powf((float)NP_, 1.0f - IW_);
    den = fmaxf(den, 1.0f);
    partial[t] = gsum[t] / den;
  }
  __syncthreads();
  if (t == 0) {
    float tot = 0.0f;
    #pragma unroll
    for (int k = 0; k < C_; ++k) tot += partial[k];
    *out = -tot / (float)(N_ * C_);
  }
}

// ---------------------------------------------------------------------------
extern "C" void kernel_launch(void* const* d_in, const int* in_sizes, int n_in,
                              void* d_out, int out_size, void* d_ws, size_t ws_size,
                              hipStream_t stream) {
  (void)in_sizes; (void)n_in; (void)out_size; (void)ws_size;
  const float*  x    = (const float*)d_in[0];
  float*        gsum = (float*)d_ws;
  unsigned int* gcnt = (unsigned int*)((float*)d_ws + C_);

  msiw_zero_ws<<<1, 64, 0, stream>>>((float*)d_ws);
  msiw_pass1<<<GRID_, BLOCK_, 0, stream>>>(x, gsum, gcnt);
  msiw_finalize<<<1, 32, 0, stream>>>(gsum, gcnt, (float*)d_out);
}